// OptNetEq_35759897707192
// MI455X (gfx1250) — compile-verified
//
#include <hip/hip_runtime.h>

// ---------------------------------------------------------------------------
// OptNetEq interior-point QP solver for MI455X (gfx1250).
// One workgroup per batch element (128 WGs x 256 threads = 8 wave32).
// Schur complement M = (A*diag(Dinv))*A^T with V_WMMA_F32_16X16X4_F32,
// exploiting symmetry (24 of 32 strips computed, lower-left mirrored).
// A stored k-permuted so one ds_load_b128 yields operands for TWO k-blocks;
// all B-operand addresses are immediate offsets off a single base register.
// 32-wide A-slabs staged in LDS shared by all 8 waves. Dense 249x249 solve
// in LDS via row-parallel Gauss-Jordan.
// ---------------------------------------------------------------------------

typedef float v2f __attribute__((ext_vector_type(2)));
typedef float v4f __attribute__((ext_vector_type(4)));
typedef float v8f __attribute__((ext_vector_type(8)));

#define QPEN   0.1f
#define SIGMA_ 0.1f
#define BIGF   1e9f
#define NN     729      // decision variables
#define MM     249      // equality constraints
#define NPAD   736      // padded n  (= 23 slabs of 32)
#define MPAD   256      // padded m  (16 tiles of 16)
#define MSTR   251      // LDS row stride for M (odd -> conflict-free columns)
#define NITER  25
#define NTHREADS 256
#define KSLAB  32       // k-columns per staged slab
#define NSLAB  23       // NPAD / KSLAB
#define ASTR   36       // slab row stride (16B rows; lr*36 mod 64 covers all
                        // 16 four-bank groups -> conflict-free b128)
#define TILEB  (16 * ASTR)   // float offset between consecutive B tiles (576)

#if defined(__has_builtin)
#if __has_builtin(__builtin_amdgcn_wmma_f32_16x16x4_f32)
#define HAVE_WMMA_F32 1
#endif
#endif

// k-permutation within each 16-group: natural q = kb<<2 | hf<<1 | r
// stored   pq = hf<<3 | kb<<1 | r   (per-lane-half data contiguous)
__host__ __device__ __forceinline__ int permk(int k) {
    int q = k & 15;
    return (k & ~15) | ((q & 2) << 2) | ((q >> 2) << 1) | (q & 1);
}
__host__ __device__ __forceinline__ int invpermk(int sk) {
    int pq = sk & 15;
    return (sk & ~15) | ((pq & 6) << 1) | ((pq & 8) >> 2) | (pq & 1);
}

// LDS layout (floats); all v4f-accessed offsets are multiples of 4 floats
#define OFF_M     0
#define OFF_Z     (MPAD * MSTR)            // 64256
#define OFF_LAM   (OFF_Z    + NPAD)
#define OFF_P     (OFF_LAM  + NPAD)
#define OFF_DINV  (OFF_P    + NPAD)        // 66464 (mult of 4)
#define OFF_RHS   (OFF_DINV + NPAD)
#define OFF_T1    (OFF_RHS  + NPAD)
#define OFF_DZ    (OFF_T1   + NPAD)
#define OFF_DLAM  (OFF_DZ   + NPAD)
#define OFF_B     (OFF_DLAM + NPAD)        // m-vectors (256 each)
#define OFF_NU    (OFF_B    + MPAD)
#define OFF_RNU   (OFF_NU   + MPAD)
#define OFF_DNU   (OFF_RNU  + MPAD)
#define OFF_RED   (OFF_DNU  + MPAD)
#define OFF_ASH   (OFF_RED  + NTHREADS)    // 71424 (mult of 4)
#define SMEM_FLOATS (OFF_ASH + MPAD * ASTR)  // 80640 floats = 322560 bytes

__device__ __forceinline__ float block_reduce_sum(float v, float* red) {
    const int t = threadIdx.x;
    red[t] = v;
    __syncthreads();
#pragma unroll
    for (int s = NTHREADS / 2; s > 0; s >>= 1) {
        if (t < s) red[t] += red[t + s];
        __syncthreads();
    }
    float r = red[0];
    __syncthreads();
    return r;
}

__device__ __forceinline__ float block_reduce_min(float v, float* red) {
    const int t = threadIdx.x;
    red[t] = v;
    __syncthreads();
#pragma unroll
    for (int s = NTHREADS / 2; s > 0; s >>= 1) {
        if (t < s) red[t] = fminf(red[t], red[t + s]);
        __syncthreads();
    }
    float r = red[0];
    __syncthreads();
    return r;
}

// Zero-padded, k-permuted copy of A: Apad[256][736].
__global__ void pad_A_kernel(const float* __restrict__ Ag,
                             float* __restrict__ Apad) {
    int idx = blockIdx.x * blockDim.x + threadIdx.x;
    if (idx >= MPAD * NPAD) return;
    int r  = idx / NPAD;
    int sk = idx - r * NPAD;
    int k  = invpermk(sk);
    Apad[idx] = (r < MM && k < NN) ? Ag[(size_t)r * NN + k] : 0.0f;
}

__global__ void __launch_bounds__(NTHREADS, 1)
optnet_ip_kernel(const float* __restrict__ puzzles,
                 const float* __restrict__ Apad,
                 const float* __restrict__ logz0,
                 float* __restrict__ out) {
    extern __shared__ float smem[];
    float* Msh   = smem + OFF_M;
    float* z_s   = smem + OFF_Z;
    float* lam_s = smem + OFF_LAM;
    float* p_s   = smem + OFF_P;
    float* dinv_s= smem + OFF_DINV;   // stored k-PERMUTED
    float* rhs_s = smem + OFF_RHS;
    float* t1_s  = smem + OFF_T1;
    float* dz_s  = smem + OFF_DZ;
    float* dlam_s= smem + OFF_DLAM;
    float* b_s   = smem + OFF_B;
    float* nu_s  = smem + OFF_NU;
    float* rnu_s = smem + OFF_RNU;
    float* dnu_s = smem + OFF_DNU;
    float* red   = smem + OFF_RED;
    float* ash   = smem + OFF_ASH;    // k-permuted slab

    const int bid  = blockIdx.x;
    const int t    = threadIdx.x;
    const int lane = t & 31;
    const int wave = t >> 5;
    const int half = lane >> 4;   // 0: lanes 0-15, 1: lanes 16-31
    const int lr   = lane & 15;

    // ---- init (natural order for all n-vectors; dinv permuted, zeroed) ----
    for (int k = t; k < NPAD; k += NTHREADS) {
        if (k < NN) {
            p_s[k]  = -puzzles[(size_t)bid * NN + k];
            t1_s[k] = __expf(logz0[k]);   // exp(log_z0) for b = A@exp(log_z0)
            z_s[k] = 1.0f; lam_s[k] = 1.0f;
        } else {
            p_s[k]  = 0.0f; t1_s[k] = 0.0f;
            z_s[k] = 0.0f; lam_s[k] = 0.0f;
        }
        dinv_s[k] = 0.0f; rhs_s[k] = 0.0f; dz_s[k] = 0.0f; dlam_s[k] = 0.0f;
    }
    __syncthreads();

    // ---- b = A @ exp(log_z0), nu = 0 ----
    if (t < MPAD) {
        float s = 0.0f;
        if (t < MM) {
            const float* Arow = Apad + (size_t)t * NPAD;
            for (int k = 0; k < NN; ++k) s += Arow[permk(k)] * t1_s[k];
        }
        b_s[t] = s;
        nu_s[t] = 0.0f;
    }
    __syncthreads();

    for (int iter = 0; iter < NITER; ++iter) {
        // ---- (1) Dinv (permuted store); mu = mean(z*lam) ----
        float zl = 0.0f;
        for (int k = t; k < NN; k += NTHREADS) {
            float zz = z_s[k], ll = lam_s[k];
            dinv_s[permk(k)] = 1.0f / (QPEN + ll / zz);
            zl += zz * ll;
        }
        float mu = block_reduce_sum(zl, red) * (1.0f / (float)NN);

        // ---- (2) rhs_z = -r_d - r_c/z ;  t1 = Dinv*rhs_z ----
        for (int k = t; k < NN; k += NTHREADS) {
            const int col = permk(k);
            float atn = 0.0f;                       // (A^T nu)[k]
            for (int i = 0; i < MM; ++i) atn += Apad[(size_t)i * NPAD + col] * nu_s[i];
            float zz = z_s[k], ll = lam_s[k];
            float rd = QPEN * zz + p_s[k] + atn - ll;
            float rc = zz * ll - SIGMA_ * mu;
            float rz = -rd - rc / zz;
            rhs_s[k] = rz;
            t1_s[k]  = dinv_s[col] * rz;
        }
        __syncthreads();

        // ---- (3) r_p = A z - b ; rhs_nu = A t1 + r_p ----
        if (t < MM) {
            const float* Arow = Apad + (size_t)t * NPAD;
            float s1 = 0.0f, s2 = 0.0f;
            for (int k = 0; k < NN; ++k) {
                float a = Arow[permk(k)];
                s1 += a * z_s[k];
                s2 += a * t1_s[k];
            }
            rnu_s[t] = s2 + (s1 - b_s[t]);
        }

        // ---- (4) M = (A*diag(Dinv)) * A^T via WMMA f32 16x16x4 ----
        // Symmetry: compute 24 strips = {TI 0-7 x both TJ-groups} u
        // {TI 8-15 x TJ-group 1}; mirror rows 128.. cols 0..127 afterwards.
#ifdef HAVE_WMMA_F32
        for (int g = 0; g < 3; ++g) {
            const int p   = wave + 8 * g;     // 0..23
            const int TI  = (p < 16) ? (p >> 1) : (p - 8);
            const int TJ0 = (p < 16) ? ((p & 1) * 8) : 8;

            // single base registers; all tile/k offsets become immediates
            const float* aptr = ash + (TI * 16 + lr) * ASTR + half * 8;
            const float* bptr = ash + (TJ0 * 16 + lr) * ASTR + half * 8;
            const float* dptr = dinv_s + half * 8;

            v8f acc[8];
#pragma unroll
            for (int j = 0; j < 8; ++j)
                acc[j] = (v8f){0.f, 0.f, 0.f, 0.f, 0.f, 0.f, 0.f, 0.f};

            for (int s = 0; s < NSLAB; ++s) {
                __syncthreads();      // previous slab fully consumed
                // cooperative copy: Apad[:, s*32 .. +32) -> ash[256][36]
#pragma unroll
                for (int i = 0; i < 8; ++i) {
                    int idx = t + i * NTHREADS;      // 0..2047 v4f chunks
                    int r  = idx >> 3;
                    int c4 = (idx & 7) << 2;
                    *(v4f*)(ash + r * ASTR + c4) =
                        *(const v4f*)(Apad + (size_t)r * NPAD + s * KSLAB + c4);
                }
                __syncthreads();

#pragma unroll
                for (int hs = 0; hs < 4; ++hs) {     // 2 k-blocks per b128
                    const int off = (hs >> 1) * 16 + (hs & 1) * 4;
                    v4f a4 = *(const v4f*)(aptr + off);
                    v4f d4 = *(const v4f*)(dptr + off);
                    v4f s4 = a4 * d4;
                    v4f b4[8];
#pragma unroll
                    for (int j = 0; j < 8; ++j)
                        b4[j] = *(const v4f*)(bptr + j * TILEB + off);

                    v2f sLo = __builtin_shufflevector(s4, s4, 0, 1);
                    v2f sHi = __builtin_shufflevector(s4, s4, 2, 3);
#pragma unroll
                    for (int j = 0; j < 8; ++j)
                        acc[j] = __builtin_amdgcn_wmma_f32_16x16x4_f32(
                            false, sLo,
                            false, __builtin_shufflevector(b4[j], b4[j], 0, 1),
                            (short)0, acc[j], false, false);
#pragma unroll
                    for (int j = 0; j < 8; ++j)
                        acc[j] = __builtin_amdgcn_wmma_f32_16x16x4_f32(
                            false, sHi,
                            false, __builtin_shufflevector(b4[j], b4[j], 2, 3),
                            (short)0, acc[j], false, false);
                }
                dptr += KSLAB;        // advance permuted-Dinv window
            }

            // store strip to M (skip cols that would overflow stride)
            const int baseRow = TI * 16 + 8 * half;
#pragma unroll
            for (int j = 0; j < 8; ++j) {
                const int col = (TJ0 + j) * 16 + lr;
                if (col < MSTR) {
#pragma unroll
                    for (int v = 0; v < 8; ++v)
                        Msh[(size_t)(baseRow + v) * MSTR + col] = acc[j][v];
                }
            }
        }
        __syncthreads();
        // mirror lower-left block: rows 128..MM-1, cols 0..127
        for (int idx = t; idx < (MM - 128) * 128; idx += NTHREADS) {
            int i = 128 + (idx >> 7);
            int j = idx & 127;
            Msh[(size_t)i * MSTR + j] = Msh[(size_t)j * MSTR + i];
        }
#else
        for (int idx = t; idx < MM * MM; idx += NTHREADS) {
            const int i = idx / MM, j = idx % MM;
            const float* Ai = Apad + (size_t)i * NPAD;
            const float* Aj = Apad + (size_t)j * NPAD;
            float s = 0.0f;
            for (int k = 0; k < NN; ++k) {
                int c = permk(k);
                s += Ai[c] * dinv_s[c] * Aj[c];
            }
            Msh[(size_t)i * MSTR + j] = s;
        }
#endif
        __syncthreads();

        // ---- (5) augment: M[:,249] = rhs_nu ----
        if (t < MM) Msh[(size_t)t * MSTR + MM] = rnu_s[t];

        // ---- (6) Gauss-Jordan (SPD, no pivoting): dnu = M \ rhs_nu ----
        for (int k = 0; k < MM; ++k) {
            __syncthreads();
            if (t < MM && t != k) {
                float piv = Msh[(size_t)k * MSTR + k];
                float f = Msh[(size_t)t * MSTR + k] / piv;
                const float* prow = Msh + (size_t)k * MSTR;
                float* rrow = Msh + (size_t)t * MSTR;
                for (int c = k + 1; c <= MM; ++c) rrow[c] -= f * prow[c];
            }
        }
        __syncthreads();
        if (t < MM)
            dnu_s[t] = Msh[(size_t)t * MSTR + MM] / Msh[(size_t)t * MSTR + t];
        __syncthreads();

        // ---- (7) dz, dlam, fraction-to-boundary alpha ----
        float amin = BIGF;
        for (int k = t; k < NN; k += NTHREADS) {
            const int col = permk(k);
            float adn = 0.0f;                       // (A^T dnu)[k]
            for (int i = 0; i < MM; ++i) adn += Apad[(size_t)i * NPAD + col] * dnu_s[i];
            float dzv = dinv_s[col] * (rhs_s[k] - adn);
            float zz = z_s[k], ll = lam_s[k];
            float rc = zz * ll - SIGMA_ * mu;
            float dlv = (-rc - ll * dzv) / zz;
            dz_s[k] = dzv;
            dlam_s[k] = dlv;
            if (dzv < 0.0f) amin = fminf(amin, -zz / dzv);
            if (dlv < 0.0f) amin = fminf(amin, -ll / dlv);
        }
        float astep = block_reduce_min(amin, red);
        float alpha = fminf(1.0f, 0.99f * astep);

        // ---- (8) update ----
        for (int k = t; k < NN; k += NTHREADS) {
            z_s[k]   += alpha * dz_s[k];
            lam_s[k] += alpha * dlam_s[k];
        }
        if (t < MM) nu_s[t] += alpha * dnu_s[t];
        __syncthreads();
    }

    // ---- output y = z ----
    for (int k = t; k < NN; k += NTHREADS)
        out[(size_t)bid * NN + k] = z_s[k];
}

extern "C" void kernel_launch(void* const* d_in, const int* in_sizes, int n_in,
                              void* d_out, int out_size, void* d_ws, size_t ws_size,
                              hipStream_t stream) {
    const float* puzzles = (const float*)d_in[0];   // [128,9,9,9] f32
    const float* A       = (const float*)d_in[1];   // [249,729]   f32
    const float* logz0   = (const float*)d_in[2];   // [729]       f32
    float* out  = (float*)d_out;
    float* Apad = (float*)d_ws;                     // 256*736 floats = 754 KB

    const int batch = in_sizes[0] / NN;
    const size_t shmem = (size_t)SMEM_FLOATS * sizeof(float);

    pad_A_kernel<<<dim3((MPAD * NPAD + NTHREADS - 1) / NTHREADS),
                   dim3(NTHREADS), 0, stream>>>(A, Apad);

    (void)hipFuncSetAttribute((const void*)optnet_ip_kernel,
                              hipFuncAttributeMaxDynamicSharedMemorySize,
                              (int)shmem);
    optnet_ip_kernel<<<dim3(batch), dim3(NTHREADS), shmem, stream>>>(
        puzzles, Apad, logz0, out);
}